// BasicRCN_57561151701200
// MI455X (gfx1250) — compile-verified
//
#include <hip/hip_runtime.h>

// Dims from the reference: B=2, P=64, T=16, D=H=128, L=2
#define Dd 128
#define Hh 128
#define Pp 64
#define Tt 16
#define Bb 2
#define Nn 32        // B*T
#define NP 2048      // Nn*Pp
#define WMAT (Dd*Hh) // 16384 floats per 128x128 matrix

typedef __attribute__((ext_vector_type(2))) float v2f;
typedef __attribute__((ext_vector_type(8))) float v8f;

// One fp32 WMMA step: D(16x16) += A(16x4) * B(4x16), full-precision CDNA5 matrix op.
// A layout: lanes 0-15 hold row M=lane&15 with K={k0,k0+1}; lanes 16-31 K={k0+2,k0+3}.
// B layout: lanes carry column N=lane&15; v[0]=K(k0+2*half), v[1]=K(k0+1+2*half).
__device__ __forceinline__ v8f wmma4(float a0, float a1, float b0, float b1, v8f c) {
  v2f a, b;
  a[0] = a0; a[1] = a1;
  b[0] = b0; b[1] = b1;
  return __builtin_amdgcn_wmma_f32_16x16x4_f32(false, a, false, b, (short)0, c,
                                               false, false);
}

// Full K=128 dot: A rows from LDS (row-major, stride lda), B from a TRANSPOSED
// weight row (Wt[col][k]) so each step is one contiguous float2 load.
__device__ __forceinline__ v8f wmma_dot128(const float* As, int lda,
                                           const float2* bt2, int half, int m,
                                           v8f c) {
  for (int k0 = 0; k0 < 128; k0 += 4) {
    int ka = k0 + 2 * half;
    float2 bb = bt2[(k0 >> 1) + half];   // floats {ka, ka+1} of Wt[col]
    c = wmma4(As[m * lda + ka], As[m * lda + ka + 1], bb.x, bb.y, c);
  }
  return c;
}

// ---------------------------------------------------------------------------
// K0: one-time transposes of all B-side weights into ws (Wt[n][k] = W[k][n]).
__global__ void transpose_w_kernel(const float* sp_w1, const float* tm_w1,
                                   const float* it_w1, const float* nw0,
                                   const float* rw0, const float* nw1,
                                   const float* rw1, const float* ow,
                                   float* wT) {
  const float* src;
  switch (blockIdx.y) {
    case 0:  src = sp_w1;           break;
    case 1:  src = sp_w1 + WMAT;    break;
    case 2:  src = tm_w1;           break;
    case 3:  src = tm_w1 + WMAT;    break;
    case 4:  src = it_w1 + WMAT;    break;
    case 5:  src = it_w1;           break;  // prod part (rows 0..D-1)
    case 6:  src = nw0;             break;
    case 7:  src = rw0;             break;
    case 8:  src = nw1;             break;
    case 9:  src = rw1;             break;
    default: src = ow;              break;
  }
  float* dst = wT + (size_t)blockIdx.y * WMAT;
  int r = blockIdx.x;   // output row = source column
  int k = threadIdx.x;  // output col = source row
  dst[(size_t)r * Dd + k] = src[(size_t)k * Hh + r];
}

// K1: combined weights, stored TRANSPOSED:  WspT[c][a] = (sp_w2 @ fus_w[0:H])[a][c]
__global__ void prep_w_kernel(const float* sp_w2, const float* tm_w2,
                              const float* it_w2, const float* fus_w,
                              float* WspT, float* WtmT, float* WitT) {
  int a = blockIdx.x;
  int c = threadIdx.x;
  const float *w2, *fus;
  float* out;
  if (blockIdx.y == 0)      { w2 = sp_w2; fus = fus_w;           out = WspT; }
  else if (blockIdx.y == 1) { w2 = tm_w2; fus = fus_w + WMAT;    out = WtmT; }
  else                      { w2 = it_w2; fus = fus_w + 2*WMAT;  out = WitT; }
  float acc = 0.f;
  for (int bb = 0; bb < Hh; ++bb) acc += w2[a*Hh + bb] * fus[bb*Hh + c];
  out[(size_t)c*Hh + a] = acc;   // transposed store
}

// K1b: combined bias b_rel, masked counts per batch-b
__global__ void prep_misc_kernel(const float* sp_b2, const float* tm_b2,
                                 const float* it_b2, const float* fus_w,
                                 const float* fus_b, const float* mask,
                                 float* b_rel, float* maskcnt, float* validc) {
  int c = threadIdx.x;
  float acc = fus_b[c];
  for (int bb = 0; bb < Hh; ++bb) {
    acc += sp_b2[bb] * fus_w[bb*Hh + c];
    acc += tm_b2[bb] * fus_w[(Hh + bb)*Hh + c];
    acc += it_b2[bb] * fus_w[(2*Hh + bb)*Hh + c];
  }
  b_rel[c] = acc;
  if (c < Bb) {
    float s = 0.f;
    for (int j = 0; j < Pp; ++j) s += mask[c*Pp + j];
    maskcnt[c] = s;
    validc[c]  = fmaxf(s, 1.0f);
  }
}

// ---------------------------------------------------------------------------
// K2: sa/sb/ta/tb/ib = gathered_feat @ {weight slice}   (WMMA GEMM, 16-row tiles)
__global__ void feat_gemm5_kernel(const float* features, const float* wT,
                                  float* sa, float* sb, float* ta, float* tb,
                                  float* ib, float* h0) {
  __shared__ float sA[16 * Dd];
  int tid = threadIdx.x;
  int r0 = blockIdx.x * 16;
  for (int idx = tid; idx < 16 * Dd; idx += 256) {
    int mi = idx >> 7, d = idx & 127;
    int r = r0 + mi;
    int n = r >> 6, p = r & 63;
    int b = n >> 4, t = n & 15;
    float v = features[((size_t)(b*Pp + p)*Tt + t)*Dd + d];
    sA[idx] = v;
    if (blockIdx.y == 0) h0[(size_t)r*Dd + d] = v;
  }
  __syncthreads();
  const float* Wt = wT + (size_t)blockIdx.y * WMAT;
  float* out;
  switch (blockIdx.y) {
    case 0:  out = sa; break;
    case 1:  out = sb; break;
    case 2:  out = ta; break;
    case 3:  out = tb; break;
    default: out = ib; break;
  }
  int lane = tid & 31, w = tid >> 5;
  int half = lane >> 4, m = lane & 15;
  int col = w * 16 + m;
  const float2* bt2 = (const float2*)(Wt + (size_t)col * Dd);
  v8f c = {};
  c = wmma_dot128(sA, Dd, bt2, half, m, c);
#pragma unroll
  for (int r = 0; r < 8; ++r) {
    int M = r + 8*half;
    out[(size_t)(r0 + M)*Hh + col] = c[r];
  }
}

// ---------------------------------------------------------------------------
// K3a: masked ReLU j-sums for spatial & temporal branches (pure VALU)
__global__ void sptm_kernel(const float* distances, const float* mask,
                            const float* sa, const float* sb, const float* ta,
                            const float* tb, const float* sp_w1,
                            const float* sp_b1, const float* tm_b1,
                            float* Ssp, float* Stm) {
  int blk = blockIdx.x;            // n*64 + i
  int n = blk >> 6, i = blk & 63;
  int b = n >> 4, t = n & 15;
  int h = threadIdx.x;
  float sai = sa[(size_t)blk*Hh + h];
  float tai = ta[(size_t)blk*Hh + h];
  float wd  = sp_w1[2*Dd*Hh + h];
  float bsp = sp_b1[h];
  float btm = tm_b1[h];
  const float* drow = distances + ((size_t)(b*Tt + t)*Pp + i)*Pp;
  float accs = 0.f, acct = 0.f;
  for (int j = 0; j < Pp; ++j) {
    float mj = mask[b*Pp + j];
    float sv = sai + sb[(size_t)(n*Pp + j)*Hh + h] + drow[j]*wd + bsp;
    float tv = tai + tb[(size_t)(n*Pp + j)*Hh + h] + btm;
    accs += mj * fmaxf(sv, 0.f);
    acct += mj * fmaxf(tv, 0.f);
  }
  Ssp[(size_t)blk*Hh + h] = accs;
  Stm[(size_t)blk*Hh + h] = acct;
}

// ---------------------------------------------------------------------------
// K3b: interaction branch, fused per (n, i-tile); ib terms staged in LDS.
__global__ void pair_it_kernel(const float* features, const float* wT,
                               const float* it_b1, const float* mask,
                               const float* ib, float* Sit) {
  __shared__ float fN[Pp * Dd];    // 32 KB
  __shared__ float sAcc[16 * Hh];  // 8 KB
  __shared__ float ibI[16 * Hh];   // 8 KB
  __shared__ float ibJ[16 * Hh];   // 8 KB
  __shared__ float mN[Pp];
  __shared__ float bN[Hh];
  int tid = threadIdx.x;
  int n = blockIdx.x >> 2;
  int i0 = (blockIdx.x & 3) * 16;
  int b = n >> 4, t = n & 15;
  for (int idx = tid; idx < Pp * Dd; idx += 256) {
    int j = idx >> 7, d = idx & 127;
    fN[idx] = features[((size_t)(b*Pp + j)*Tt + t)*Dd + d];
  }
  for (int idx = tid; idx < 16 * Hh; idx += 256) {
    sAcc[idx] = 0.f;
    ibI[idx]  = ib[(size_t)(n*Pp + i0)*Hh + idx];
  }
  if (tid < Pp) mN[tid] = mask[b*Pp + tid];
  if (tid < Hh) bN[tid] = it_b1[tid];
  __syncthreads();
  int lane = tid & 31, w = tid >> 5;
  int half = lane >> 4, m = lane & 15;
  int col = w * 16 + m;
  float bc = bN[col];
  const float2* bt2 =
      (const float2*)(wT + 5 * (size_t)WMAT + (size_t)col * Dd);
  for (int jt = 0; jt < 4; ++jt) {
    int j0 = jt * 16;
    __syncthreads();
    for (int idx = tid; idx < 16 * Hh; idx += 256)
      ibJ[idx] = ib[(size_t)(n*Pp + j0)*Hh + idx];
    __syncthreads();
    const float* fj = &fN[(j0 + m) * Dd];
    for (int mi = 0; mi < 16; ++mi) {
      const float* fi = &fN[(i0 + mi) * Dd];
      v8f c = {};
      for (int k0 = 0; k0 < Dd; k0 += 4) {
        int ka = k0 + 2*half;
        float a0 = fi[ka]     * fj[ka];
        float a1 = fi[ka + 1] * fj[ka + 1];
        float2 bb = bt2[(k0 >> 1) + half];
        c = wmma4(a0, a1, bb.x, bb.y, c);
      }
      float ibi = ibI[mi*Hh + col];
      float partial = 0.f;
#pragma unroll
      for (int r = 0; r < 8; ++r) {
        int M = r + 8*half;
        float v = c[r] + ibi + ibJ[M*Hh + col] + bc;
        partial += fmaxf(v, 0.f) * mN[j0 + M];
      }
      partial += __shfl_xor(partial, 16, 32);
      if (lane < 16) sAcc[mi*Hh + col] += partial;
    }
  }
  __syncthreads();
  for (int idx = tid; idx < 16 * Hh; idx += 256)
    Sit[(size_t)(n*Pp + i0)*Hh + idx] = sAcc[idx];
}

// ---------------------------------------------------------------------------
// K4: rsum = m_i * (Ssp@Wsp + Stm@Wtm + Sit@Wit + cnt_b * b_rel)
__global__ void relsum_kernel(const float* Ssp, const float* Stm,
                              const float* Sit, const float* WspT,
                              const float* WtmT, const float* WitT,
                              const float* b_rel, const float* mask,
                              const float* maskcnt, float* rsum) {
  __shared__ float tS[3][16 * Hh];
  int tid = threadIdx.x;
  int r0 = blockIdx.x * 16;
  for (int idx = tid; idx < 16 * Hh; idx += 256) {
    tS[0][idx] = Ssp[(size_t)r0*Hh + idx];
    tS[1][idx] = Stm[(size_t)r0*Hh + idx];
    tS[2][idx] = Sit[(size_t)r0*Hh + idx];
  }
  __syncthreads();
  int lane = tid & 31, w = tid >> 5;
  int half = lane >> 4, m = lane & 15;
  int col = w * 16 + m;
  v8f c = {};
  c = wmma_dot128(tS[0], Hh, (const float2*)(WspT + (size_t)col*Hh), half, m, c);
  c = wmma_dot128(tS[1], Hh, (const float2*)(WtmT + (size_t)col*Hh), half, m, c);
  c = wmma_dot128(tS[2], Hh, (const float2*)(WitT + (size_t)col*Hh), half, m, c);
  float brc = b_rel[col];
#pragma unroll
  for (int r = 0; r < 8; ++r) {
    int M = r + 8*half;
    int rr = r0 + M;
    int n = rr >> 6, i = rr & 63, b = n >> 4;
    rsum[(size_t)rr*Hh + col] = (c[r] + maskcnt[b]*brc) * mask[b*Pp + i];
  }
}

// ---------------------------------------------------------------------------
// K5: one RelationConv layer
__global__ void layer_kernel(const float* hin, const float* rsum,
                             const float* nwT, const float* node_b,
                             const float* rwT, const float* rel_b,
                             const float* aggs_p, const float* ln_g,
                             const float* ln_b, const float* mask,
                             const float* validc, float* hout) {
  __shared__ float tH[16 * Hh];
  __shared__ float tR[16 * Hh];
  __shared__ float tO[16 * Hh];
  __shared__ float muS[16], rstdS[16];
  int tid = threadIdx.x;
  int r0 = blockIdx.x * 16;
  for (int idx = tid; idx < 16 * Hh; idx += 256) {
    tH[idx] = hin[(size_t)r0*Hh + idx];
    tR[idx] = rsum[(size_t)r0*Hh + idx];
  }
  __syncthreads();
  int lane = tid & 31, w = tid >> 5;
  int half = lane >> 4, m = lane & 15;
  int col = w * 16 + m;
  v8f cn = {}, cr = {};
  cn = wmma_dot128(tH, Hh, (const float2*)(nwT + (size_t)col*Hh), half, m, cn);
  cr = wmma_dot128(tR, Hh, (const float2*)(rwT + (size_t)col*Hh), half, m, cr);
  float aggs = aggs_p[0];
  float nb = node_b[col], rb = rel_b[col];
#pragma unroll
  for (int r = 0; r < 8; ++r) {
    int M = r + 8*half;
    int rr = r0 + M;
    int n = rr >> 6, i = rr & 63, b = n >> 4;
    float agg = (cr[r] + (float)Pp * rb) * mask[b*Pp + i] / validc[b];
    tO[M*Hh + col] = cn[r] + nb + aggs * agg;
  }
  __syncthreads();
  if (tid < 16) {
    float s = 0.f, s2 = 0.f;
    for (int cc = 0; cc < Hh; ++cc) {
      float v = tO[tid*Hh + cc];
      s += v; s2 += v*v;
    }
    float mean = s / (float)Hh;
    float var = s2 / (float)Hh - mean*mean;
    muS[tid] = mean;
    rstdS[tid] = rsqrtf(var + 1e-5f);
  }
  __syncthreads();
  for (int idx = tid; idx < 16 * Hh; idx += 256) {
    int M = idx >> 7, cc = idx & 127;
    float v = (tO[idx] - muS[M]) * rstdS[M] * ln_g[cc] + ln_b[cc];
    hout[(size_t)r0*Hh + idx] = fmaxf(v, 0.f);
  }
}

// ---------------------------------------------------------------------------
// K6: out = features + scatter( h@out_w + out_b )
__global__ void out_kernel(const float* hin, const float* owT,
                           const float* out_b, const float* features,
                           float* out) {
  __shared__ float tH[16 * Hh];
  int tid = threadIdx.x;
  int r0 = blockIdx.x * 16;
  for (int idx = tid; idx < 16 * Hh; idx += 256)
    tH[idx] = hin[(size_t)r0*Hh + idx];
  __syncthreads();
  int lane = tid & 31, w = tid >> 5;
  int half = lane >> 4, m = lane & 15;
  int col = w * 16 + m;
  v8f c = {};
  c = wmma_dot128(tH, Hh, (const float2*)(owT + (size_t)col*Hh), half, m, c);
  float ob = out_b[col];
#pragma unroll
  for (int r = 0; r < 8; ++r) {
    int M = r + 8*half;
    int rr = r0 + M;
    int n = rr >> 6, p = rr & 63, b = n >> 4, t = n & 15;
    size_t oi = ((size_t)(b*Pp + p)*Tt + t)*Dd + col;
    out[oi] = features[oi] + c[r] + ob;
  }
}

// ---------------------------------------------------------------------------
extern "C" void kernel_launch(void* const* d_in, const int* in_sizes, int n_in,
                              void* d_out, int out_size, void* d_ws,
                              size_t ws_size, hipStream_t stream) {
  const float* features  = (const float*)d_in[0];
  const float* distances = (const float*)d_in[1];
  const float* mask      = (const float*)d_in[2];
  const float* sp_w1 = (const float*)d_in[3];
  const float* sp_b1 = (const float*)d_in[4];
  const float* sp_w2 = (const float*)d_in[5];
  const float* sp_b2 = (const float*)d_in[6];
  const float* tm_w1 = (const float*)d_in[7];
  const float* tm_b1 = (const float*)d_in[8];
  const float* tm_w2 = (const float*)d_in[9];
  const float* tm_b2 = (const float*)d_in[10];
  const float* it_w1 = (const float*)d_in[11];
  const float* it_b1 = (const float*)d_in[12];
  const float* it_w2 = (const float*)d_in[13];
  const float* it_b2 = (const float*)d_in[14];
  const float* fus_w = (const float*)d_in[15];
  const float* fus_b = (const float*)d_in[16];
  const float* out_w = (const float*)d_in[17];
  const float* out_b = (const float*)d_in[18];
  const float* l_node_w[2]; const float* l_node_b[2];
  const float* l_rel_w[2];  const float* l_rel_b[2];
  const float* l_agg[2];    const float* l_ln_g[2]; const float* l_ln_b[2];
  for (int l = 0; l < 2; ++l) {
    int base = 19 + l * 7;
    l_node_w[l] = (const float*)d_in[base + 0];
    l_node_b[l] = (const float*)d_in[base + 1];
    l_rel_w[l]  = (const float*)d_in[base + 2];
    l_rel_b[l]  = (const float*)d_in[base + 3];
    l_agg[l]    = (const float*)d_in[base + 4];
    l_ln_g[l]   = (const float*)d_in[base + 5];
    l_ln_b[l]   = (const float*)d_in[base + 6];
  }

  float* ws = (float*)d_ws;
  size_t o = 0;
  const size_t NPH = (size_t)NP * Hh;
  float* sa   = ws + o; o += NPH;
  float* sb   = ws + o; o += NPH;
  float* ta   = ws + o; o += NPH;
  float* tb   = ws + o; o += NPH;
  float* ib   = ws + o; o += NPH;
  float* Ssp  = ws + o; o += NPH;
  float* Stm  = ws + o; o += NPH;
  float* Sit  = ws + o; o += NPH;
  float* rsum = ws + o; o += NPH;
  float* h0   = ws + o; o += NPH;
  float* h1   = ws + o; o += NPH;
  float* wT   = ws + o; o += (size_t)11 * WMAT;
  float* WspT = ws + o; o += (size_t)WMAT;
  float* WtmT = ws + o; o += (size_t)WMAT;
  float* WitT = ws + o; o += (size_t)WMAT;
  float* b_rel   = ws + o; o += Hh;
  float* maskcnt = ws + o; o += 2;
  float* validc  = ws + o; o += 2;
  (void)ws_size; (void)in_sizes; (void)n_in; (void)out_size;

  const int ROWTILES = NP / 16;  // 128

  transpose_w_kernel<<<dim3(Hh, 11), Dd, 0, stream>>>(
      sp_w1, tm_w1, it_w1, l_node_w[0], l_rel_w[0], l_node_w[1], l_rel_w[1],
      out_w, wT);
  prep_w_kernel<<<dim3(Hh, 3), Hh, 0, stream>>>(sp_w2, tm_w2, it_w2, fus_w,
                                                WspT, WtmT, WitT);
  prep_misc_kernel<<<1, Hh, 0, stream>>>(sp_b2, tm_b2, it_b2, fus_w, fus_b,
                                         mask, b_rel, maskcnt, validc);
  feat_gemm5_kernel<<<dim3(ROWTILES, 5), 256, 0, stream>>>(
      features, wT, sa, sb, ta, tb, ib, h0);
  sptm_kernel<<<NP, Hh, 0, stream>>>(distances, mask, sa, sb, ta, tb, sp_w1,
                                     sp_b1, tm_b1, Ssp, Stm);
  pair_it_kernel<<<Nn * 4, 256, 0, stream>>>(features, wT, it_b1, mask, ib,
                                             Sit);
  relsum_kernel<<<ROWTILES, 256, 0, stream>>>(Ssp, Stm, Sit, WspT, WtmT, WitT,
                                              b_rel, mask, maskcnt, rsum);
  layer_kernel<<<ROWTILES, 256, 0, stream>>>(
      h0, rsum, wT + 6*(size_t)WMAT, l_node_b[0], wT + 7*(size_t)WMAT,
      l_rel_b[0], l_agg[0], l_ln_g[0], l_ln_b[0], mask, validc, h1);
  layer_kernel<<<ROWTILES, 256, 0, stream>>>(
      h1, rsum, wT + 8*(size_t)WMAT, l_node_b[1], wT + 9*(size_t)WMAT,
      l_rel_b[1], l_agg[1], l_ln_g[1], l_ln_b[1], mask, validc, h0);
  out_kernel<<<ROWTILES, 256, 0, stream>>>(h0, wT + 10*(size_t)WMAT, out_b,
                                           features, (float*)d_out);
}